// S4Layer_1090921693698
// MI455X (gfx1250) — compile-verified
//
#include <hip/hip_runtime.h>
#include <hip/hip_bf16.h>

#ifndef __has_builtin
#define __has_builtin(x) 0
#endif

#if __has_builtin(__builtin_amdgcn_tensor_load_to_lds) && \
    __has_builtin(__builtin_amdgcn_s_wait_tensorcnt)
#define USE_TDM 1
#else
#define USE_TDM 0
#endif

#define DMODEL   256
#define NSTATE   16
#define BATCH    8
#define SEQLEN   2048
#define CHUNK    128                   // phase-1/2 chunk length
#define NCHUNK   (SEQLEN / CHUNK)      // 16
#define TILE     16                    // WMMA tile (timesteps)
#define TILES_PER_CHUNK (CHUNK / TILE) // 8

typedef float v2f __attribute__((ext_vector_type(2)));
typedef float v8f __attribute__((ext_vector_type(8)));
typedef unsigned int u32x4 __attribute__((ext_vector_type(4)));
typedef int i32x4v __attribute__((ext_vector_type(4)));
typedef int i32x8v __attribute__((ext_vector_type(8)));

static __device__ __forceinline__ v2f make2(float a, float b) {
  v2f r; r.x = a; r.y = b; return r;
}

static __device__ __forceinline__ float sigm(float v) {
  return 1.0f / (1.0f + __expf(-v));
}

// ---------------------------------------------------------------------------
// Phase 1: per (b, d, chunk) compute the chunk-local end state with h0 = 0.
// Coalesced x reads (lanes map to consecutive d). 32768 threads.
// ---------------------------------------------------------------------------
__global__ void __launch_bounds__(256) s4_phase1(
    const float* __restrict__ x, const float* __restrict__ A,
    const float* __restrict__ Bm, const float* __restrict__ delta,
    float* __restrict__ hloc) {
  int tid  = blockIdx.x * blockDim.x + threadIdx.x;   // 0..32767
  int d    = tid & (DMODEL - 1);
  int rest = tid >> 8;                                // 0..127
  int c    = rest & (NCHUNK - 1);
  int b    = rest >> 4;                               // 0..7

  float dt = sigm(delta[d]);
  float abar[NSTATE], bbar[NSTATE], h[NSTATE];
#pragma unroll
  for (int n = 0; n < NSTATE; n++) {
    abar[n] = __expf(dt * A[d * NSTATE + n]);
    bbar[n] = dt * Bm[d * NSTATE + n];
    h[n]    = 0.0f;
  }

  const float* xp = x + ((size_t)(b * SEQLEN + c * CHUNK)) * DMODEL + d;
  int tbase = c * CHUNK;
  for (int t = 0; t < CHUNK; t++) {
    if ((t & 31) == 0 && (tbase + t + 64) < SEQLEN)
      __builtin_prefetch(xp + (size_t)(t + 64) * DMODEL, 0, 1);  // global_prefetch_b8
    float xv = xp[(size_t)t * DMODEL];
#pragma unroll
    for (int n = 0; n < NSTATE; n++)
      h[n] = fmaf(abar[n], h[n], bbar[n] * xv);
  }

  float* out = hloc + ((size_t)(b * DMODEL + d) * NCHUNK + c) * NSTATE;
#pragma unroll
  for (int n = 0; n < NSTATE; n++) out[n] = h[n];
}

// ---------------------------------------------------------------------------
// Phase 2: prefix-combine chunk states: hin[c] = Abar^128 * hin[c-1] + hloc[c-1].
// One thread per (b, d, n): 16-step serial combine. 32768 threads.
// ---------------------------------------------------------------------------
__global__ void __launch_bounds__(256) s4_phase2(
    const float* __restrict__ A, const float* __restrict__ delta,
    const float* __restrict__ hloc, float* __restrict__ hin) {
  int tid = blockIdx.x * blockDim.x + threadIdx.x;    // 0..32767
  int n = tid & (NSTATE - 1);
  int d = (tid >> 4) & (DMODEL - 1);
  int b = tid >> 12;                                  // 0..7

  float dt = sigm(delta[d]);
  float a  = __expf(dt * A[d * NSTATE + n]);
  float a128 = a;
#pragma unroll
  for (int i = 0; i < 7; i++) a128 *= a128;           // a^128

  size_t base = ((size_t)(b * DMODEL + d) * NCHUNK) * NSTATE + n;
  float h = 0.0f;
  for (int c = 0; c < NCHUNK; c++) {
    hin[base + (size_t)c * NSTATE] = h;               // state at chunk start
    h = fmaf(a128, h, hloc[base + (size_t)c * NSTATE]);
  }
}

// ---------------------------------------------------------------------------
// Phase 3: one wave per (d, chunk). TDM gathers the strided X tile
// (1 d x 128 t x 8 b, strides 1KB / 2MB) into LDS, overlapped with the
// per-d WMMA operand construction. Then 8 sequential 16-step tiles:
//   Y    = M_d @ X  +  G_d @ Hin          (8x v_wmma_f32_16x16x4_f32)
//   Hloc = W_d @ X                        (4x v_wmma_f32_16x16x4_f32)
//   Hin  = diag(Abar^16) Hin + Hloc       (ds_swizzle SWAPX16 + VALU)
// Columns 0..7 = batch, 8..15 = padding (zero).
// ---------------------------------------------------------------------------
__global__ void __launch_bounds__(32) s4_phase3(
    const float* __restrict__ x, const float* __restrict__ A,
    const float* __restrict__ Bm, const float* __restrict__ Cm,
    const float* __restrict__ Dv, const float* __restrict__ delta,
    const float* __restrict__ hin, float* __restrict__ y) {
  int d = blockIdx.x & (DMODEL - 1);
  int c = blockIdx.x >> 8;              // 0..15
  int lane = threadIdx.x;               // 0..31
  int half = lane >> 4;                 // 0 = K rows 4k..4k+1, 1 = 4k+2..4k+3
  int row  = lane & 15;                 // A-operand row (t or n); B/C/D column b

#if USE_TDM
  __shared__ __align__(16) float xtile[CHUNK * BATCH];  // [b*128 + t], 4 KB

  // --- Tensor DMA descriptor (CDNA5 ISA): tile (1, 128, 8) of
  // f32 from x[b, c*128 + t, d]; dim0 = d (contig), dim1 = t (stride 256),
  // dim2 = b (stride 2048*256). All sizes/strides in data_size units.
  {
    unsigned long long ga =
        (unsigned long long)(uintptr_t)(x + ((size_t)(c * CHUNK)) * DMODEL + d);
    unsigned lds_off = (unsigned)(uintptr_t)&xtile[0];

    u32x4 g0;
    g0.x = 1u;                                      // count=1, user descriptor
    g0.y = lds_off;                                 // lds_addr [63:32]
    g0.z = (unsigned)(ga & 0xFFFFFFFFu);            // global_addr lo
    g0.w = (unsigned)((ga >> 32) & 0x01FFFFFFu)     // global_addr hi (57b)
         | 0x80000000u;                             // type=2 ("image") [127:126]

    i32x8v g1;
    g1[0] = 0x00020000;                 // wg_mask=0, data_size=2 (4B)
    g1[1] = (int)(256u << 16);          // tensor_dim0=256  (bits 79:48, lo16)
    g1[2] = (int)(2048u << 16);         // tensor_dim1=2048 (bits 111:80, lo16)
    g1[3] = 0x00010000;                 // tile_dim0=1      (bits 127:112)
    g1[4] = (int)((8u << 16) | 128u);   // tile_dim1=128, tile_dim2=8
    g1[5] = 256;                        // tensor_dim0_stride = 256
    g1[6] = 0;                          // stride0 hi16 | stride1 lo16 (0x80000 -> 0)
    g1[7] = (int)(524288u >> 16);       // tensor_dim1_stride = 2048*256 (hi bits)

    i32x4v g2; g2[0] = 8; g2[1] = 0; g2[2] = 0; g2[3] = 0;  // tensor_dim2=8
    i32x4v g3; g3[0] = 0; g3[1] = 0; g3[2] = 0; g3[3] = 0;

#if __has_include(<hip/amd_detail/amd_gfx1250_TDM.h>)
    // amdgpu-toolchain (clang-23): 6-arg form
    i32x8v g4; g4[0]=0; g4[1]=0; g4[2]=0; g4[3]=0; g4[4]=0; g4[5]=0; g4[6]=0; g4[7]=0;
    __builtin_amdgcn_tensor_load_to_lds(g0, g1, g2, g3, g4, 0);
#else
    // ROCm 7.2 (clang-22): 5-arg form
    __builtin_amdgcn_tensor_load_to_lds(g0, g1, g2, g3, 0);
#endif
  }
#endif  // USE_TDM

  float dt = sigm(delta[d]);
  float abar[NSTATE], bbar[NSTATE], Cv[NSTATE];
#pragma unroll
  for (int n = 0; n < NSTATE; n++) {
    abar[n] = __expf(dt * A[d * NSTATE + n]);
    bbar[n] = dt * Bm[d * NSTATE + n];
    Cv[n]   = Cm[d * NSTATE + n];
  }

  // Impulse response K[tau] = sum_n Cv*bbar*abar^tau (registers only).
  float K[16], q[16];
#pragma unroll
  for (int n = 0; n < NSTATE; n++) q[n] = Cv[n] * bbar[n];
#pragma unroll
  for (int tau = 0; tau < 16; tau++) {
    float s = 0.0f;
#pragma unroll
    for (int n = 0; n < NSTATE; n++) s += q[n];
    K[tau] = s;
    if (tau < 15) {
#pragma unroll
      for (int n = 0; n < NSTATE; n++) q[n] *= abar[n];
    }
  }

  // Row t=row of lower-triangular Toeplitz M_d (select chain, no scratch).
  int t = row;
  float Mrow[16];
#pragma unroll
  for (int s = 0; s < 16; s++) {
    float v = 0.0f;
#pragma unroll
    for (int tau = 0; tau < 16; tau++) v = ((t - s) == tau) ? K[tau] : v;
    Mrow[s] = (s <= t) ? v : 0.0f;
  }

  // Row t of carry matrix G_d: grow[n] = Cv[n] * abar[n]^(t+1).
  float grow[16];
#pragma unroll
  for (int n = 0; n < NSTATE; n++) {
    float p = abar[n];
    for (int i = 0; i < t; i++) p *= abar[n];
    grow[n] = Cv[n] * p;
  }

  // Row n=row of state matrix W_d: wrow[s] = abar_n^(15-s) * bbar_n.
  float a_n, b_n;
  {
    float an = 0.0f, bn = 0.0f;
#pragma unroll
    for (int n = 0; n < NSTATE; n++) {
      an = (row == n) ? abar[n] : an;
      bn = (row == n) ? bbar[n] : bn;
    }
    a_n = an; b_n = bn;
  }
  float pj[16];
  pj[0] = 1.0f;
#pragma unroll
  for (int j = 1; j < 16; j++) pj[j] = pj[j - 1] * a_n;
  float wrow[16];
#pragma unroll
  for (int s = 0; s < 16; s++) wrow[s] = pj[15 - s] * b_n;

  // A-operand slices (ISA 16x4 layout: lanes0-15 -> K=4k,4k+1; lanes16-31 -> K=4k+2,4k+3)
  v2f mA[4], gA[4], wA[4];
#pragma unroll
  for (int k = 0; k < 4; k++) {
    mA[k] = half ? make2(Mrow[4 * k + 2], Mrow[4 * k + 3]) : make2(Mrow[4 * k], Mrow[4 * k + 1]);
    gA[k] = half ? make2(grow[4 * k + 2], grow[4 * k + 3]) : make2(grow[4 * k], grow[4 * k + 1]);
    wA[k] = half ? make2(wrow[4 * k + 2], wrow[4 * k + 3]) : make2(wrow[4 * k], wrow[4 * k + 1]);
  }

  // Abar^16 (diagonal tile-to-tile decay).
  float a16[16];
#pragma unroll
  for (int n = 0; n < NSTATE; n++) {
    float p = abar[n];
    p *= p; p *= p; p *= p; p *= p;
    a16[n] = p;
  }

  // Replicated per-column state: rows 0..7 in h0[], rows 8..15 in h1[].
  int b = row;
  bool act = (b < BATCH);
  float h0[8], h1[8];
  if (act) {
    const float* hp = hin + ((size_t)(b * DMODEL + d) * NCHUNK + c) * NSTATE;
#pragma unroll
    for (int r = 0; r < 8; r++) { h0[r] = hp[r]; h1[r] = hp[8 + r]; }
  } else {
#pragma unroll
    for (int r = 0; r < 8; r++) { h0[r] = 0.0f; h1[r] = 0.0f; }
  }
  float Dd = Dv[d];

#if USE_TDM
  __builtin_amdgcn_s_wait_tensorcnt(0);   // X tile resident in LDS
#endif

  for (int st = 0; st < TILES_PER_CHUNK; st++) {
    int t0 = c * CHUNK + st * TILE;

    // B-operand X slices: lane supplies column b, K rows 4k+2*half + {0,1}.
    v2f xb[4];
#pragma unroll
    for (int k = 0; k < 4; k++) {
      if (act) {
        int s0 = 4 * k + 2 * half;
#if USE_TDM
        xb[k] = *(const v2f*)&xtile[b * CHUNK + st * TILE + s0];  // ds_load_b64
#else
        size_t i0 = ((size_t)(b * SEQLEN + t0 + s0)) * DMODEL + d;
        xb[k] = make2(x[i0], x[i0 + DMODEL]);
#endif
      } else {
        xb[k] = make2(0.0f, 0.0f);
      }
    }

    // Y = M_d @ X  (chained K=16 via 4x 16x16x4)
    v8f Y;
#pragma unroll
    for (int r = 0; r < 8; r++) Y[r] = 0.0f;
#pragma unroll
    for (int k = 0; k < 4; k++)
      Y = __builtin_amdgcn_wmma_f32_16x16x4_f32(false, mA[k], false, xb[k],
                                                (short)0, Y, false, false);

    // Y += G_d @ Hin  (Hin slices selected in-lane from replicated state)
    v2f hb[4];
    hb[0] = half ? make2(h0[2], h0[3]) : make2(h0[0], h0[1]);
    hb[1] = half ? make2(h0[6], h0[7]) : make2(h0[4], h0[5]);
    hb[2] = half ? make2(h1[2], h1[3]) : make2(h1[0], h1[1]);
    hb[3] = half ? make2(h1[6], h1[7]) : make2(h1[4], h1[5]);
#pragma unroll
    for (int k = 0; k < 4; k++)
      Y = __builtin_amdgcn_wmma_f32_16x16x4_f32(false, gA[k], false, hb[k],
                                                (short)0, Y, false, false);

    // Hloc = W_d @ X
    v8f Hl;
#pragma unroll
    for (int r = 0; r < 8; r++) Hl[r] = 0.0f;
#pragma unroll
    for (int k = 0; k < 4; k++)
      Hl = __builtin_amdgcn_wmma_f32_16x16x4_f32(false, wA[k], false, xb[k],
                                                 (short)0, Hl, false, false);

    // D-layout: lane holds rows 8*half + r of its column; partner lane (^16)
    // holds the other 8 rows. One SWAPX16 ds_swizzle per register.
    float own[8], oth[8];
#pragma unroll
    for (int r = 0; r < 8; r++) {
      own[r] = Hl[r];
      oth[r] = __int_as_float(
          __builtin_amdgcn_ds_swizzle(__float_as_int(Hl[r]), 0x401F));
    }
#pragma unroll
    for (int r = 0; r < 8; r++) {
      float lo = half ? oth[r] : own[r];  // rows 0..7
      float hi = half ? own[r] : oth[r];  // rows 8..15
      h0[r] = fmaf(a16[r],     h0[r], lo);
      h1[r] = fmaf(a16[8 + r], h1[r], hi);
    }

    // Epilogue: y = Y + D*x (only real batch columns).
    if (act) {
#pragma unroll
      for (int r = 0; r < 8; r++) {
        int tt = t0 + 8 * half + r;
        size_t idx = ((size_t)(b * SEQLEN + tt)) * DMODEL + d;
#if USE_TDM
        float xv = xtile[b * CHUNK + st * TILE + 8 * half + r];
#else
        float xv = x[idx];
#endif
        y[idx] = Y[r] + Dd * xv;
      }
    }
  }
}

// ---------------------------------------------------------------------------
extern "C" void kernel_launch(void* const* d_in, const int* in_sizes, int n_in,
                              void* d_out, int out_size, void* d_ws, size_t ws_size,
                              hipStream_t stream) {
  const float* x     = (const float*)d_in[0];
  const float* A     = (const float*)d_in[1];
  const float* Bm    = (const float*)d_in[2];
  const float* Cm    = (const float*)d_in[3];
  const float* Dv    = (const float*)d_in[4];
  const float* delta = (const float*)d_in[5];
  float* y = (float*)d_out;

  float* hloc = (float*)d_ws;                                    // 2 MiB
  float* hin  = hloc + (size_t)BATCH * DMODEL * NCHUNK * NSTATE; // 2 MiB

  s4_phase1<<<128, 256, 0, stream>>>(x, A, Bm, delta, hloc);
  s4_phase2<<<128, 256, 0, stream>>>(A, delta, hloc, hin);
  s4_phase3<<<DMODEL * NCHUNK, 32, 0, stream>>>(x, A, Bm, Cm, Dv, delta, hin, y);
}